// SimpleRuleEnhancedTransH_69020124446752
// MI455X (gfx1250) — compile-verified
//
#include <hip/hip_runtime.h>
#include <hip/hip_bf16.h>

typedef float v2f __attribute__((ext_vector_type(2)));
typedef float v8f __attribute__((ext_vector_type(8)));

#define DIM      512
#define NPOS     1024
#define NNEG     8192
#define NRULE    20
#define NREL     500
#define NRULEPAD 32

// ---------------- workspace layout (floats) ----------------
#define WS_WN     0                         // [NREL*DIM]      normalized hyperplanes
#define WS_B      (WS_WN + NREL*DIM)        // [512]           b_r = re . wn
#define WS_RN2    (WS_B + 512)              // [512]           ||re||^2
#define WS_DIFF   (WS_RN2 + 512)            // [NPOS*DIM]      he - te per pos triple
#define WS_ND2    (WS_DIFF + NPOS*DIM)      // [NPOS]          ||diff||^2
#define WS_DPOS   (WS_ND2 + NPOS)           // [NPOS]
#define WS_DNEG   (WS_DPOS + NPOS)          // [NNEG]
#define WS_ACC    (WS_DNEG + NNEG)          // [1]             rule-loss accumulator

__device__ __forceinline__ float waveReduceSum(float v) {
    #pragma unroll
    for (int m = 16; m > 0; m >>= 1) v += __shfl_xor(v, m, 32);
    return v;
}

// valid result in wave 0 (all its lanes); ends with a barrier so sbuf is reusable
__device__ __forceinline__ float blockSum256(float v, float* sbuf) {
    v = waveReduceSum(v);
    const int wid  = threadIdx.x >> 5;
    const int lane = threadIdx.x & 31;
    if (lane == 0) sbuf[wid] = v;
    __syncthreads();
    float t = (threadIdx.x < 8) ? sbuf[threadIdx.x] : 0.0f;
    if (wid == 0) t = waveReduceSum(t);
    __syncthreads();
    return t;
}

__global__ void k_init(float* acc) {
    if (threadIdx.x == 0) acc[0] = 0.0f;
}

// one block per relation: wn = w/||w||, b = re.wn, rn2 = ||re||^2
__global__ void k_prep(const float* __restrict__ norm_vec,
                       const float* __restrict__ rel_emb,
                       float* __restrict__ wn, float* __restrict__ bvals,
                       float* __restrict__ rn2) {
    const int r   = blockIdx.x;
    const int tid = threadIdx.x;
    __shared__ float sbuf[8];
    __shared__ float s_inv;

    const float* w  = norm_vec + (long)r * DIM;
    const float* re = rel_emb  + (long)r * DIM;
    float e0 = w[tid], e1 = w[tid + 256];
    float nrm = blockSum256(e0 * e0 + e1 * e1, sbuf);
    if (tid == 0) s_inv = rsqrtf(fmaxf(nrm, 1e-30f));
    __syncthreads();
    const float inv = s_inv;
    float w0 = e0 * inv, w1 = e1 * inv;
    wn[(long)r * DIM + tid]       = w0;
    wn[(long)r * DIM + tid + 256] = w1;

    float r0 = re[tid], r1 = re[tid + 256];
    float bsum = blockSum256(r0 * w0 + r1 * w1, sbuf);
    float rsum = blockSum256(r0 * r0 + r1 * r1, sbuf);
    if (tid == 0) { bvals[r] = bsum; rn2[r] = rsum; }
}

// one block per pos triple: diff row + ||diff||^2
__global__ void k_diff(const int* __restrict__ pos,
                       const float* __restrict__ ent,
                       float* __restrict__ diff, float* __restrict__ nd2) {
    const int i   = blockIdx.x;
    const int tid = threadIdx.x;
    __shared__ float sbuf[8];
    const int h = pos[3 * i + 0];
    const int t = pos[3 * i + 2];
    const float* he = ent + (long)h * DIM;
    const float* te = ent + (long)t * DIM;
    float d0 = he[tid] - te[tid];
    float d1 = he[tid + 256] - te[tid + 256];
    diff[(long)i * DIM + tid]       = d0;
    diff[(long)i * DIM + tid + 256] = d1;
    float s = blockSum256(d0 * d0 + d1 * d1, sbuf);
    if (tid == 0) nd2[i] = s;
}

// one wave per distance: 1024 pos + 8192 neg TransH distances
__global__ void k_dist(const int* __restrict__ pos, const int* __restrict__ neg,
                       const float* __restrict__ ent, const float* __restrict__ rel,
                       const float* __restrict__ wn,
                       float* __restrict__ dpos, float* __restrict__ dneg) {
    const int lane = threadIdx.x & 31;
    const int wave = blockIdx.x * (blockDim.x >> 5) + (threadIdx.x >> 5);
    if (wave >= NPOS + NNEG) return;
    const int* trip = (wave < NPOS) ? (pos + 3 * wave) : (neg + 3 * (wave - NPOS));
    const int h = trip[0], r = trip[1], t = trip[2];

    const float4* he = (const float4*)(ent + (long)h * DIM + lane * 16);
    const float4* te = (const float4*)(ent + (long)t * DIM + lane * 16);
    const float4* re = (const float4*)(rel + (long)r * DIM + lane * 16);
    const float4* w  = (const float4*)(wn  + (long)r * DIM + lane * 16);

    float4 H[4], T[4], R[4], W[4];
    #pragma unroll
    for (int q = 0; q < 4; ++q) { H[q] = he[q]; T[q] = te[q]; R[q] = re[q]; W[q] = w[q]; }
    const float* hf = (const float*)H;
    const float* tf = (const float*)T;
    const float* rf = (const float*)R;
    const float* wf = (const float*)W;

    float dot = 0.0f;
    #pragma unroll
    for (int e = 0; e < 16; ++e) dot += (hf[e] - tf[e]) * wf[e];
    dot = waveReduceSum(dot);          // c = (he-te) . w

    float s = 0.0f;
    #pragma unroll
    for (int e = 0; e < 16; ++e) {
        float v = (hf[e] - tf[e]) + rf[e] - dot * wf[e];
        s += v * v;
    }
    s = waveReduceSum(s);
    if (lane == 0) {
        float d = sqrtf(s);
        if (wave < NPOS) dpos[wave] = d; else dneg[wave - NPOS] = d;
    }
}

// WMMA rule kernel: one wave per 16-row tile of Diff; N = 32 (rules padded),
// K = 512. Two GEMMs (Diff x Wn^T, Diff x Re^T) in fp32 via v_wmma_f32_16x16x4_f32.
__global__ void k_rule(const int* __restrict__ pos, const int* __restrict__ r1,
                       const int* __restrict__ r2, const float* __restrict__ conf,
                       const float* __restrict__ rel_emb, const float* __restrict__ wn,
                       const float* __restrict__ bvals, const float* __restrict__ rn2,
                       const float* __restrict__ diff, const float* __restrict__ nd2,
                       float* __restrict__ acc) {
    const int lane  = threadIdx.x;         // blockDim.x == 32, EXEC all ones
    const int m0    = blockIdx.x * 16;
    const int row   = lane & 15;
    const int khalf = (lane >> 4) * 2;     // A/B fragment K sub-offset (0 or 2)

    // B columns: two N-tiles of 16; cols >= NRULE are padding (computed, never read)
    const int col0 = row;
    const int col1 = 16 + row;
    const int rr0  = (col0 < NRULE) ? r2[col0] : 0;
    const int rr1  = (col1 < NRULE) ? r2[col1] : 0;

    const float* pA  = diff    + (long)(m0 + row) * DIM + khalf;
    const float* pW0 = wn      + (long)rr0 * DIM + khalf;
    const float* pW1 = wn      + (long)rr1 * DIM + khalf;
    const float* pR0 = rel_emb + (long)rr0 * DIM + khalf;
    const float* pR1 = rel_emb + (long)rr1 * DIM + khalf;

    v8f cw0 = {}, cw1 = {}, cr0 = {}, cr1 = {};
    #pragma unroll 4
    for (int k0 = 0; k0 < DIM; k0 += 4) {
        __builtin_prefetch(pA + k0 + 128, 0, 0);
        v2f a  = *(const v2f*)(pA  + k0);
        v2f b0 = *(const v2f*)(pW0 + k0);
        v2f b1 = *(const v2f*)(pW1 + k0);
        v2f b2 = *(const v2f*)(pR0 + k0);
        v2f b3 = *(const v2f*)(pR1 + k0);
        cw0 = __builtin_amdgcn_wmma_f32_16x16x4_f32(false, a, false, b0, (short)0, cw0, false, false);
        cw1 = __builtin_amdgcn_wmma_f32_16x16x4_f32(false, a, false, b1, (short)0, cw1, false, false);
        cr0 = __builtin_amdgcn_wmma_f32_16x16x4_f32(false, a, false, b2, (short)0, cr0, false, false);
        cr1 = __builtin_amdgcn_wmma_f32_16x16x4_f32(false, a, false, b3, (short)0, cr1, false, false);
    }

    // Epilogue: lane holds C rows M = v + 8*(lane>>4), col N = lane&15 per tile.
    const int jl  = lane & 15;
    const int mhi = 8 * (lane >> 4);
    float lsum = 0.0f;
    #pragma unroll
    for (int v = 0; v < 8; ++v) {
        const int i  = m0 + v + mhi;
        const int ri = pos[3 * i + 1];
        const float nd = nd2[i];
        if (jl < NRULE) {                 // tile 0: cols 0..15 always < 20
            const int   j  = jl;
            const int   rr = rr0;
            const float c1 = cw0[v], c2 = cr0[v];
            float d2 = nd + rn2[rr] + 2.0f * c2 - c1 * c1 - 2.0f * c1 * bvals[rr];
            float d  = sqrtf(fmaxf(d2, 0.0f));
            if (ri == r1[j]) lsum += conf[j] * d;
        }
        if (16 + jl < NRULE) {            // tile 1: cols 16..19 valid
            const int   j  = 16 + jl;
            const int   rr = rr1;
            const float c1 = cw1[v], c2 = cr1[v];
            float d2 = nd + rn2[rr] + 2.0f * c2 - c1 * c1 - 2.0f * c1 * bvals[rr];
            float d  = sqrtf(fmaxf(d2, 0.0f));
            if (ri == r1[j]) lsum += conf[j] * d;
        }
    }
    lsum = waveReduceSum(lsum);
    if (lane == 0) atomicAdd(acc, lsum);
}

// final: mean(relu(1 + dpos[k/8] - dneg[k])) + 0.5 * rule_sum
__global__ void k_final(const float* __restrict__ dpos, const float* __restrict__ dneg,
                        const float* __restrict__ acc, float* __restrict__ out) {
    const int tid = threadIdx.x;
    __shared__ float sbuf[8];
    float s = 0.0f;
    for (int k = tid; k < NNEG; k += 256)
        s += fmaxf(0.0f, 1.0f + dpos[k >> 3] - dneg[k]);
    s = blockSum256(s, sbuf);
    if (tid == 0) out[0] = s * (1.0f / (float)NNEG) + 0.5f * acc[0];
}

extern "C" void kernel_launch(void* const* d_in, const int* in_sizes, int n_in,
                              void* d_out, int out_size, void* d_ws, size_t ws_size,
                              hipStream_t stream) {
    const int*   pos  = (const int*)d_in[0];
    const int*   neg  = (const int*)d_in[1];
    const int*   r1   = (const int*)d_in[2];
    const int*   r2   = (const int*)d_in[3];
    const float* conf = (const float*)d_in[4];
    const float* ent  = (const float*)d_in[5];
    const float* rel  = (const float*)d_in[6];
    const float* nv   = (const float*)d_in[7];
    float* out = (float*)d_out;

    float* ws    = (float*)d_ws;
    float* wn    = ws + WS_WN;
    float* bvals = ws + WS_B;
    float* rn2   = ws + WS_RN2;
    float* diff  = ws + WS_DIFF;
    float* nd2   = ws + WS_ND2;
    float* dpos  = ws + WS_DPOS;
    float* dneg  = ws + WS_DNEG;
    float* acc   = ws + WS_ACC;

    k_init<<<1, 32, 0, stream>>>(acc);
    k_prep<<<NREL, 256, 0, stream>>>(nv, rel, wn, bvals, rn2);
    k_diff<<<NPOS, 256, 0, stream>>>(pos, ent, diff, nd2);
    k_dist<<<(NPOS + NNEG) / 8, 256, 0, stream>>>(pos, neg, ent, rel, wn, dpos, dneg);
    k_rule<<<NPOS / 16, 32, 0, stream>>>(pos, r1, r2, conf, rel, wn, bvals, rn2, diff, nd2, acc);
    k_final<<<1, 256, 0, stream>>>(dpos, dneg, acc, out);
}